// LFMAAdapter_75608604278844
// MI455X (gfx1250) — compile-verified
//
#include <hip/hip_runtime.h>
#include <hip/hip_bf16.h>

#define D1 4096
#define D2 4096
#define NROWS 128          // 2*64 rows of x
#define ALPHA 16.0f

typedef float v2f __attribute__((ext_vector_type(2)));
typedef float v8f __attribute__((ext_vector_type(8)));

// ---------------------------------------------------------------------------
// 1) zero the complex frequency buffer F (128 MB) with float4 stores
// ---------------------------------------------------------------------------
__global__ void zero_f4(float4* __restrict__ p, unsigned int n4) {
    unsigned int i = blockIdx.x * blockDim.x + threadIdx.x;
    unsigned int stride = gridDim.x * blockDim.x;
    float4 z = make_float4(0.f, 0.f, 0.f, 0.f);
    for (; i < n4; i += stride) p[i] = z;
}

// ---------------------------------------------------------------------------
// 2) scatter sparse complex coefficients (indices are a permutation -> unique)
// ---------------------------------------------------------------------------
__global__ void scatter_coef(const float* __restrict__ cr,
                             const float* __restrict__ ci,
                             const int* __restrict__ idx,
                             float2* __restrict__ F, int K) {
    int k = blockIdx.x * blockDim.x + threadIdx.x;
    if (k < K) F[(size_t)idx[k]] = make_float2(cr[k], ci[k]);
}

// ---------------------------------------------------------------------------
// 4096-point radix-2 DIT inverse FFT in LDS (sign +, unnormalized here)
// ---------------------------------------------------------------------------
__device__ __forceinline__ void ifft4096_stages(float2* s, int tid) {
#pragma unroll 1
    for (int st = 1; st <= 12; ++st) {
        const int half = 1 << (st - 1);
#pragma unroll
        for (int b = tid; b < 2048; b += 512) {
            const int grp = b >> (st - 1);
            const int j   = b & (half - 1);
            const int i0  = (grp << st) + j;
            const int i1  = i0 + half;
            float sn, cs;
            sincospif((float)j / (float)half, &sn, &cs);   // theta = +2*pi*j/len
            float2 u = s[i0];
            float2 v = s[i1];
            float2 t = make_float2(v.x * cs - v.y * sn, v.x * sn + v.y * cs);
            s[i0] = make_float2(u.x + t.x, u.y + t.y);
            s[i1] = make_float2(u.x - t.x, u.y - t.y);
        }
        __syncthreads();
    }
}

// 3) in-place row IFFT of F, scaled by 1/4096
__global__ __launch_bounds__(512) void ifft_rows_c2c(float2* __restrict__ F) {
    __shared__ float2 s[D2];
    const int tid = threadIdx.x;
    float2* row = F + (size_t)blockIdx.x * D2;
    for (int i = tid; i < D2; i += 512) s[__brev((unsigned)i) >> 20] = row[i];
    __syncthreads();
    ifft4096_stages(s, tid);
    const float inv = 1.0f / 4096.0f;
    for (int i = tid; i < D2; i += 512) {
        float2 v = s[i];
        row[i] = make_float2(v.x * inv, v.y * inv);
    }
}

// 5) row IFFT of G (= columns of original), emit real part * (ALPHA/4096)
//    into DW_T[f][d]  (transposed Delta_W, same layout as W_base)
__global__ __launch_bounds__(512) void ifft_rows_real(const float2* __restrict__ G,
                                                      float* __restrict__ DWt) {
    __shared__ float2 s[D1];
    const int tid = threadIdx.x;
    const float2* row = G + (size_t)blockIdx.x * D1;
    for (int i = tid; i < D1; i += 512) s[__brev((unsigned)i) >> 20] = row[i];
    __syncthreads();
    ifft4096_stages(s, tid);
    const float scale = ALPHA / 4096.0f;
    float* orow = DWt + (size_t)blockIdx.x * D1;
    for (int i = tid; i < D1; i += 512) orow[i] = s[i].x * scale;
}

// ---------------------------------------------------------------------------
// 4) coalesced complex transpose via padded LDS tile
// ---------------------------------------------------------------------------
__global__ __launch_bounds__(256) void transpose_c(const float2* __restrict__ in,
                                                   float2* __restrict__ out) {
    __shared__ float2 tile[32][33];
    const int tx = threadIdx.x, ty = threadIdx.y;
    const int c0 = blockIdx.x * 32, r0 = blockIdx.y * 32;
#pragma unroll
    for (int j = 0; j < 32; j += 8)
        tile[ty + j][tx] = in[(size_t)(r0 + ty + j) * D2 + c0 + tx];
    __syncthreads();
#pragma unroll
    for (int j = 0; j < 32; j += 8)
        out[(size_t)(c0 + ty + j) * D1 + r0 + tx] = tile[tx][ty + j];
}

// ---------------------------------------------------------------------------
// 6) fused GEMM: out[m,f] = sum_d x[m,d]*(Wb[f,d]+DWt[f,d]) + bias[f]
//    256 WGs (one per 16-wide f tile) x 8 waves (one per 16-row m tile).
//    K chunked by 64 through LDS (stride-68 padding -> conflict-free
//    float2 fragment reads), V_WMMA_F32_16X16X4_F32 accumulation.
// ---------------------------------------------------------------------------
__global__ __launch_bounds__(256) void gemm_wmma(const float* __restrict__ x,
                                                 const float* __restrict__ Wb,
                                                 const float* __restrict__ bb,
                                                 const float* __restrict__ DWt,
                                                 float* __restrict__ out) {
    __shared__ float As[NROWS * 68];   // 128 x 64 chunk of x
    __shared__ float Bs[16 * 68];      // 16  x 64 chunk of (Wb + DWt)

    const int tid   = threadIdx.x;
    const int f0    = blockIdx.x * 16;
    const int wave  = tid >> 5;
    const int lane  = tid & 31;
    const int n     = lane & 15;       // M index for A frag, N index for B/C
    const int hi    = lane >> 4;
    const int mbase = wave * 16;

    v8f acc = {};

    for (int kc = 0; kc < D1; kc += 64) {
        // ---- cooperative, coalesced loads into LDS ----
#pragma unroll
        for (int i = 0; i < 8; ++i) {
            int t   = tid + i * 256;        // 2048 float4 total
            int row = t >> 4;
            int c4  = (t & 15) << 2;
            *(float4*)&As[row * 68 + c4] =
                *(const float4*)&x[(size_t)row * D1 + kc + c4];
        }
        {
            int row = tid >> 4;             // 256 float4 total
            int c4  = (tid & 15) << 2;
            const size_t off = (size_t)(f0 + row) * D1 + kc + c4;
            float4 w = *(const float4*)&Wb[off];
            float4 d = *(const float4*)&DWt[off];
            float4 sv = make_float4(w.x + d.x, w.y + d.y, w.z + d.z, w.w + d.w);
            *(float4*)&Bs[row * 68 + c4] = sv;
        }
        __syncthreads();

        // ---- 16 WMMA steps of K=4 each ----
#pragma unroll
        for (int i = 0; i < 16; ++i) {
            const int k0 = 4 * i + 2 * hi;
            v2f a = *(const v2f*)&As[(mbase + n) * 68 + k0];  // A[m][k0..k0+1]
            v2f b = *(const v2f*)&Bs[n * 68 + k0];            // B[k0..k0+1][n]
            acc = __builtin_amdgcn_wmma_f32_16x16x4_f32(
                false, a, false, b, (short)0, acc, false, false);
        }
        __syncthreads();
    }

    const float bias = bb[f0 + n];
#pragma unroll
    for (int v = 0; v < 8; ++v) {
        const int m = mbase + v + 8 * hi;                     // C layout: M=v+8*hi
        out[(size_t)m * D2 + f0 + n] = acc[v] + bias;
    }
}

// ---------------------------------------------------------------------------
extern "C" void kernel_launch(void* const* d_in, const int* in_sizes, int n_in,
                              void* d_out, int out_size, void* d_ws, size_t ws_size,
                              hipStream_t stream) {
    const float* x    = (const float*)d_in[0];
    const float* Wb   = (const float*)d_in[1];
    const float* bb   = (const float*)d_in[2];
    const float* cr   = (const float*)d_in[3];
    const float* ci   = (const float*)d_in[4];
    const int*   midx = (const int*)d_in[5];
    float*       out  = (float*)d_out;
    const int K = in_sizes[3];

    char* ws = (char*)d_ws;
    float2* F   = (float2*)ws;                                  // 128 MB
    float2* G   = (float2*)(ws + (size_t)128 * 1024 * 1024);    // 128 MB
    float*  DWt = (float*)ws;                                   // reuses F (dead)

    // 1) zero F
    zero_f4<<<8192, 256, 0, stream>>>((float4*)F,
                                      (unsigned)((size_t)D1 * D2 * 2 / 4));
    // 2) scatter coefficients
    scatter_coef<<<(K + 255) / 256, 256, 0, stream>>>(cr, ci, midx, F, K);
    // 3) row IFFT (axis 1), in place, x 1/4096
    ifft_rows_c2c<<<D1, 512, 0, stream>>>(F);
    // 4) transpose F -> G
    transpose_c<<<dim3(D2 / 32, D1 / 32), dim3(32, 8), 0, stream>>>(F, G);
    // 5) column IFFT via rows of G, real part * (ALPHA/4096) -> DW_T[f][d]
    ifft_rows_real<<<D2, 512, 0, stream>>>(G, DWt);
    // 6) fused WMMA GEMM
    gemm_wmma<<<D2 / 16, 256, 0, stream>>>(x, Wb, bb, DWt, out);
}